// FractionalPooling_54150947668745
// MI455X (gfx1250) — compile-verified
//
#include <hip/hip_runtime.h>

typedef float v2f __attribute__((ext_vector_type(2)));
typedef float v8f __attribute__((ext_vector_type(8)));

#define GL_ALPHA   0.5f
#define WLEN       256      // row length (last axis)
#define HDIM       256
#define OUT_W      128
#define ROWS_WG    16
#define LDS_STRIDE 260      // 256 + 4 pad: stride%64==4 -> conflict-free column reads

// ---------------------------------------------------------------------------
// Kernel 1: build T[n][m] = w[n-m] for m<=n else 0  (row-major in n) so a
// WMMA B-fragment lane (fixed n, consecutive k) is one contiguous b64 load.
// ---------------------------------------------------------------------------
__global__ void gl_build_T(float* __restrict__ T) {
    __shared__ float w[WLEN];
    const int t = threadIdx.x;
    if (t == 0) {
        float acc = 1.0f;
        w[0] = 1.0f;
        for (int k = 1; k < WLEN; ++k) {
            acc *= ((float)k - 1.0f - GL_ALPHA) / (float)k;
            w[k] = acc;
        }
    }
    __syncthreads();
    const int n = t;
    float* row = T + (long)n * WLEN;
    for (int m = 0; m < WLEN; ++m)
        row[m] = (m <= n) ? w[n - m] : 0.0f;
}

// ---------------------------------------------------------------------------
// Fragment helpers: 4 consecutive K-steps (K=16) worth of fragments.
// ---------------------------------------------------------------------------
__device__ __forceinline__ void load4(v2f d[4], const float* base, int k) {
#pragma unroll
    for (int i = 0; i < 4; ++i) d[i] = *(const v2f*)(base + k + 4 * i);
}

__device__ __forceinline__ void copy4(v2f d[4], const v2f s[4]) {
#pragma unroll
    for (int i = 0; i < 4; ++i) d[i] = s[i];
}

__device__ __forceinline__ void wmma4(v8f& c, const v2f a[4], const v2f b[4]) {
#pragma unroll
    for (int i = 0; i < 4; ++i)
        c = __builtin_amdgcn_wmma_f32_16x16x4_f32(
                false, a[i], false, b[i], (short)0, c, false, false);
}

// 2x2 max-pool of one 16x16 C/D tile and store the pooled 8x8 block.
__device__ __forceinline__ void pool_store(const v8f& c, int nbase, int h0,
                                           int bc, int lane, int llow,
                                           int lhalf, float* __restrict__ out) {
    // C/D layout: VGPR v holds M=v (lanes 0-15) / M=v+8 (lanes 16-31), N=llow.
    float r0 = fmaxf(c[0], c[1]);
    float r1 = fmaxf(c[2], c[3]);
    float r2 = fmaxf(c[4], c[5]);
    float r3 = fmaxf(c[6], c[7]);
    // column pairs live in adjacent lanes
    float m0 = fmaxf(r0, __shfl_xor(r0, 1, 32));
    float m1 = fmaxf(r1, __shfl_xor(r1, 1, 32));
    float m2 = fmaxf(r2, __shfl_xor(r2, 1, 32));
    float m3 = fmaxf(r3, __shfl_xor(r3, 1, 32));
    if ((lane & 1) == 0) {
        const int pc = (nbase >> 1) + (llow >> 1);   // pooled col 0..127
        const int pr = (h0 >> 1) + lhalf * 4;        // pooled row base
        float* o = out + (long)bc * (OUT_W * (HDIM / 2))
                       + (long)pr * OUT_W + pc;
        o[0 * OUT_W] = m0;
        o[1 * OUT_W] = m1;
        o[2 * OUT_W] = m2;
        o[3 * OUT_W] = m3;
    }
}

// ---------------------------------------------------------------------------
// Kernel 2: per-WG: 16x256 X tile -> LDS; wave w computes n-blocks {w, 15-w}.
// Phase 1 (k < kLo): one shared A fragment feeds two WMMA accumulator chains.
// Phase 2 (kLo <= k < kHi): finish the longer (jHi) block.
// 16-K unrolled bodies with next-tile loads issued before current WMMAs.
// ---------------------------------------------------------------------------
__global__ __launch_bounds__(256)
void gl_frac_pool(const float* __restrict__ X, const float* __restrict__ T,
                  float* __restrict__ out) {
    __shared__ float Xs[ROWS_WG * LDS_STRIDE];

    const int tid = threadIdx.x;
    const long rowBase = (long)blockIdx.x * ROWS_WG;

    // ---- cooperative load: 16 rows x 256 f32 -> LDS ----
    {
        const int r  = tid >> 4;
        const int c0 = (tid & 15) * 16;
        const float4* src = (const float4*)(X + (rowBase + r) * WLEN + c0);
#pragma unroll
        for (int i = 0; i < 4; ++i) {
            float4 v = src[i];
            *(float4*)(&Xs[r * LDS_STRIDE + c0 + i * 4]) = v;
        }
    }
    __syncthreads();

    const int wave  = tid >> 5;
    const int lane  = tid & 31;
    const int lhalf = lane >> 4;          // selects K pair {0,1} vs {2,3}
    const int llow  = lane & 15;

    const int bc = (int)(rowBase / HDIM);
    const int h0 = (int)(rowBase % HDIM);

    const float* Xrow = &Xs[llow * LDS_STRIDE + 2 * lhalf];

    const int jLo = wave;                 // triangular pairing: w and 15-w
    const int jHi = 15 - wave;
    // scalar (SGPR) loop bounds -> uniform s_cbranch loops, EXEC stays all-1s
    const int kLo = __builtin_amdgcn_readfirstlane((jLo + 1) * 16);
    const int kHi = __builtin_amdgcn_readfirstlane((jHi + 1) * 16);

    const float* TLo = T + (long)(jLo * 16 + llow) * WLEN + 2 * lhalf;
    const float* THi = T + (long)(jHi * 16 + llow) * WLEN + 2 * lhalf;

    v8f cLo = {};
    v8f cHi = {};
    v2f a[4], an[4], bl[4], bln[4], bh[4], bhn[4];

    // ================= phase 1: shared A, two accumulator chains ==========
    load4(a,  Xrow, 0);
    load4(bl, TLo,  0);
    load4(bh, THi,  0);
    for (int k = 0; k + 16 < kLo; k += 16) {
        load4(an,  Xrow, k + 16);         // prefetch next 16-K group
        load4(bln, TLo,  k + 16);
        load4(bhn, THi,  k + 16);
        wmma4(cLo, a, bl);                // 8 WMMAs overlap the loads above
        wmma4(cHi, a, bh);
        copy4(a, an); copy4(bl, bln); copy4(bh, bhn);
    }
    wmma4(cLo, a, bl);
    wmma4(cHi, a, bh);

    // ================= phase 2: finish jHi block ==========================
    load4(a,  Xrow, kLo);
    load4(bh, THi,  kLo);
    for (int k = kLo; k + 16 < kHi; k += 16) {
        load4(an,  Xrow, k + 16);
        load4(bhn, THi,  k + 16);
        wmma4(cHi, a, bh);
        copy4(a, an); copy4(bh, bhn);
    }
    wmma4(cHi, a, bh);

    // ================= pool + store =======================================
    pool_store(cLo, jLo * 16, h0, bc, lane, llow, lhalf, out);
    pool_store(cHi, jHi * 16, h0, bc, lane, llow, lhalf, out);
}

// ---------------------------------------------------------------------------
extern "C" void kernel_launch(void* const* d_in, const int* in_sizes, int n_in,
                              void* d_out, int out_size, void* d_ws, size_t ws_size,
                              hipStream_t stream) {
    (void)in_sizes; (void)n_in; (void)out_size; (void)ws_size;
    const float* x = (const float*)d_in[0];
    float* out     = (float*)d_out;
    float* T       = (float*)d_ws;                 // 256*256*4 = 256 KB scratch

    hipLaunchKernelGGL(gl_build_T, dim3(1), dim3(WLEN), 0, stream, T);

    const long R = (long)16 * 64 * HDIM;           // 262144 rows
    hipLaunchKernelGGL(gl_frac_pool, dim3((unsigned)(R / ROWS_WG)), dim3(256),
                       0, stream, x, T, out);
}